// DPLRSSMLayer_62302795596503
// MI455X (gfx1250) — compile-verified
//
#include <hip/hip_runtime.h>

typedef __attribute__((ext_vector_type(2))) float v2f;
typedef __attribute__((ext_vector_type(8))) float v8f;

namespace {
constexpr int  kD      = 512;
constexpr int  kL      = 4096;

// workspace layout (float offsets)
constexpr long OFF_W1 = 0;                   // 512x128 paired-K  (65536)
constexpr long OFF_W2 = OFF_W1 + 65536;      // 128x512 paired-K  (65536)
constexpr long OFF_MS = OFF_W2 + 65536;      // M^64: 64x64 re | im (8192)
constexpr long OFF_Z  = OFF_MS + 8192;       // chunk terminal states re|im (131072)
constexpr long OFF_XS = OFF_Z  + 131072;     // chunk initial states re|im (131072)
constexpr long OFF_X  = OFF_XS + 131072;     // Xin then X: 65536x128 (8388608)
}

// ---------------------------------------------------------------------------
// prep: coefficients, W1 (B_bar panel), W2 (C panel), M^64 via LDS squaring
// ---------------------------------------------------------------------------
__global__ void prep_kernel(const float* __restrict__ lnr, const float* __restrict__ im,
                            const float* __restrict__ Pr,  const float* __restrict__ Pi,
                            const float* __restrict__ Qr,  const float* __restrict__ Qi,
                            const float* __restrict__ Br,  const float* __restrict__ Bi,
                            const float* __restrict__ Cr,  const float* __restrict__ Ci,
                            const float* __restrict__ logdt, float* __restrict__ ws)
{
    __shared__ float s_cr[64], s_ci[64], s_ar[64], s_ai[64];
    __shared__ float m_r[4096], m_i[4096];
    const int tid = threadIdx.x;                 // 256 threads
    const float dt = expf(logdt[0]);

    if (tid < 64) {
        float lr_ = -expf(lnr[tid]);
        float li_ = im[tid];
        float e   = expf(lr_ * dt);
        float abr = e * cosf(li_ * dt);
        float abi = e * sinf(li_ * dt);
        float dr = lr_ + 1e-8f, di = li_;
        float den = dr*dr + di*di;
        float nr = abr - 1.0f, ni = abi;
        s_ar[tid] = abr;  s_ai[tid] = abi;
        s_cr[tid] = (nr*dr + ni*di) / den * dt;
        s_ci[tid] = (ni*dr - nr*di) / den * dt;
    }
    __syncthreads();

    // W1: paired-K, float pair p=k/2 at OFF_W1 + 2*(p*128 + n)
    for (int slot = tid; slot < 32768; slot += 256) {
        int kp = slot >> 7, n = slot & 127, k = kp * 2;
        float v0, v1;
        if (n < 64) {
            float cr = s_cr[n], ci = s_ci[n];
            v0 = cr*Br[n*kD + k]   - ci*Bi[n*kD + k];
            v1 = cr*Br[n*kD + k+1] - ci*Bi[n*kD + k+1];
        } else {
            int m = n - 64;
            float cr = s_cr[m], ci = s_ci[m];
            v0 = cr*Bi[m*kD + k]   + ci*Br[m*kD + k];
            v1 = cr*Bi[m*kD + k+1] + ci*Br[m*kD + k+1];
        }
        ws[OFF_W1 + 2*slot]     = v0;
        ws[OFF_W1 + 2*slot + 1] = v1;
    }

    // W2: K=128 (n re | -n im), N=512 (d); paired-K
    for (int slot = tid; slot < 32768; slot += 256) {
        int kp = slot >> 9, d = slot & 511, k = kp * 2;
        float v0, v1;
        if (k < 64) { v0 =  Cr[d*64 + k];      v1 =  Cr[d*64 + k + 1]; }
        else        { v0 = -Ci[d*64 + k - 64]; v1 = -Ci[d*64 + k - 63]; }
        ws[OFF_W2 + 2*slot]     = v0;
        ws[OFF_W2 + 2*slot + 1] = v1;
    }

    // M = diag(A_bar) + dt * P * Q^H
    for (int e = tid; e < 4096; e += 256) {
        int i = e >> 6, j = e & 63;
        float vr = 0.f, vi = 0.f;
#pragma unroll
        for (int r = 0; r < 4; r++) {
            float pr = Pr[i*4+r], pi = Pi[i*4+r];
            float qr = Qr[j*4+r], qi = -Qi[j*4+r];
            vr += pr*qr - pi*qi;
            vi += pr*qi + pi*qr;
        }
        vr *= dt; vi *= dt;
        if (i == j) { vr += s_ar[i]; vi += s_ai[i]; }
        m_r[e] = vr; m_i[e] = vi;
    }
    __syncthreads();

    // M^64 via 6 squarings, in place in LDS
    for (int it = 0; it < 6; it++) {
        float outr[16], outi[16];
#pragma unroll
        for (int s = 0; s < 16; s++) {
            int e = tid + s*256, i = e >> 6, j = e & 63;
            float vr = 0.f, vi = 0.f;
            for (int kq = 0; kq < 64; kq++) {
                float ar = m_r[i*64+kq], ai = m_i[i*64+kq];
                float br = m_r[kq*64+j], bi = m_i[kq*64+j];
                vr += ar*br - ai*bi;
                vi += ar*bi + ai*br;
            }
            outr[s] = vr; outi[s] = vi;
        }
        __syncthreads();
#pragma unroll
        for (int s = 0; s < 16; s++) { int e = tid + s*256; m_r[e] = outr[s]; m_i[e] = outi[s]; }
        __syncthreads();
    }
    for (int e = tid; e < 4096; e += 256) {
        ws[OFF_MS + e]        = m_r[e];
        ws[OFF_MS + 4096 + e] = m_i[e];
    }
}

// ---------------------------------------------------------------------------
// GEMM1: Xin(65536x128) = u(65536x512) @ W1(512x128), fp32 WMMA 16x16x4.
// Each wave computes a 32x64 strip: 2 A fragments share 4 B fragments
// (6 b64 loads per 8 WMMAs).
// ---------------------------------------------------------------------------
__global__ void gemm1_kernel(const float* __restrict__ u, float* __restrict__ ws)
{
    const int lane  = threadIdx.x & 31;
    const int wave  = blockIdx.x * (blockDim.x >> 5) + (threadIdx.x >> 5);
    const int mt    = wave >> 1;          // 32-row strip index (0..2047)
    const int n0    = (wave & 1) * 64;
    const int lr    = lane & 15;
    const int khalf = (lane >> 4) << 1;   // 0 or 2
    const int hi8   = (lane >> 4) << 3;   // 0 or 8
    const float* W1 = ws + OFF_W1;
    float* X        = ws + OFF_X;
    const long r0   = (long)mt * 32;
    const float* uptr0 = u + (r0 + lr) * kD;
    const float* uptr1 = u + (r0 + 16 + lr) * kD;

    v8f a00 = {}, a01 = {}, a02 = {}, a03 = {};
    v8f a10 = {}, a11 = {}, a12 = {}, a13 = {};
    for (int k = 0; k < kD; k += 4) {
        int kk = k + khalf;
        v2f av0 = *(const v2f*)(uptr0 + kk);
        v2f av1 = *(const v2f*)(uptr1 + kk);
        const float* wb = W1 + (long)kk * 128 + 2 * (n0 + lr);
        v2f b0 = *(const v2f*)(wb);
        v2f b1 = *(const v2f*)(wb + 32);
        v2f b2 = *(const v2f*)(wb + 64);
        v2f b3 = *(const v2f*)(wb + 96);
        a00 = __builtin_amdgcn_wmma_f32_16x16x4_f32(false, av0, false, b0, (short)0, a00, false, false);
        a01 = __builtin_amdgcn_wmma_f32_16x16x4_f32(false, av0, false, b1, (short)0, a01, false, false);
        a02 = __builtin_amdgcn_wmma_f32_16x16x4_f32(false, av0, false, b2, (short)0, a02, false, false);
        a03 = __builtin_amdgcn_wmma_f32_16x16x4_f32(false, av0, false, b3, (short)0, a03, false, false);
        a10 = __builtin_amdgcn_wmma_f32_16x16x4_f32(false, av1, false, b0, (short)0, a10, false, false);
        a11 = __builtin_amdgcn_wmma_f32_16x16x4_f32(false, av1, false, b1, (short)0, a11, false, false);
        a12 = __builtin_amdgcn_wmma_f32_16x16x4_f32(false, av1, false, b2, (short)0, a12, false, false);
        a13 = __builtin_amdgcn_wmma_f32_16x16x4_f32(false, av1, false, b3, (short)0, a13, false, false);
    }
#pragma unroll
    for (int v = 0; v < 8; v++) {
        float* op0 = X + (r0 + hi8 + v) * 128 + n0 + lr;
        float* op1 = X + (r0 + 16 + hi8 + v) * 128 + n0 + lr;
        op0[0]  = a00[v];  op0[16] = a01[v];  op0[32] = a02[v];  op0[48] = a03[v];
        op1[0]  = a10[v];  op1[16] = a11[v];  op1[32] = a12[v];  op1[48] = a13[v];
    }
}

// ---------------------------------------------------------------------------
// chunked scan: one wave per (batch, chunk); lane owns states n and n+32.
// WRITE_X=false: local scan from zero, emit terminal state Z.
// WRITE_X=true : scan from Xstart, overwrite Xin rows with state trajectory.
// ---------------------------------------------------------------------------
template <bool WRITE_X>
__global__ void scan_kernel(const float* __restrict__ lnr, const float* __restrict__ im,
                            const float* __restrict__ Pr,  const float* __restrict__ Pi,
                            const float* __restrict__ Qr,  const float* __restrict__ Qi,
                            const float* __restrict__ logdt, float* __restrict__ ws)
{
    const int wid  = blockIdx.x;     // 0..1023
    const int b    = wid >> 6;
    const int c    = wid & 63;
    const int lane = threadIdx.x;    // 0..31
    const int n0 = lane, n1 = lane + 32;
    const float dt = expf(logdt[0]);

    float l0r = -expf(lnr[n0]), l0i = im[n0];
    float e0  = expf(l0r * dt);
    float a0r = e0 * cosf(l0i * dt), a0i = e0 * sinf(l0i * dt);
    float l1r = -expf(lnr[n1]), l1i = im[n1];
    float e1  = expf(l1r * dt);
    float a1r = e1 * cosf(l1i * dt), a1i = e1 * sinf(l1i * dt);

    float p0r[4], p0i[4], p1r[4], p1i[4], q0r[4], q0i[4], q1r[4], q1i[4];
#pragma unroll
    for (int r = 0; r < 4; r++) {
        p0r[r] = Pr[n0*4+r];  p0i[r] = Pi[n0*4+r];
        p1r[r] = Pr[n1*4+r];  p1i[r] = Pi[n1*4+r];
        q0r[r] = Qr[n0*4+r];  q0i[r] = -Qi[n0*4+r];   // conj(Q)
        q1r[r] = Qr[n1*4+r];  q1i[r] = -Qi[n1*4+r];
    }

    float x0r = 0.f, x0i = 0.f, x1r = 0.f, x1i = 0.f;
    if (WRITE_X) {
        long si = (long)(b*64 + c) * 64;
        x0r = ws[OFF_XS + si + n0];          x0i = ws[OFF_XS + 65536 + si + n0];
        x1r = ws[OFF_XS + si + n1];          x1i = ws[OFF_XS + 65536 + si + n1];
    }

    float* X = ws + OFF_X;
    const long base = ((long)b * kL + (long)c * 64) * 128;
    for (int t = 0; t < 64; t++) {
        float* xp = X + base + (long)t * 128;
        float f0r = xp[n0],      f0i = xp[64 + n0];
        float f1r = xp[n1],      f1i = xp[64 + n1];

        float qxr[4], qxi[4];
#pragma unroll
        for (int r = 0; r < 4; r++) {
            float srv = q0r[r]*x0r - q0i[r]*x0i + q1r[r]*x1r - q1i[r]*x1i;
            float siv = q0r[r]*x0i + q0i[r]*x0r + q1r[r]*x1i + q1i[r]*x1r;
#pragma unroll
            for (int m = 16; m >= 1; m >>= 1) {
                srv += __shfl_xor(srv, m, 32);
                siv += __shfl_xor(siv, m, 32);
            }
            qxr[r] = srv; qxi[r] = siv;
        }

        float pq0r = 0.f, pq0i = 0.f, pq1r = 0.f, pq1i = 0.f;
#pragma unroll
        for (int r = 0; r < 4; r++) {
            pq0r += p0r[r]*qxr[r] - p0i[r]*qxi[r];
            pq0i += p0r[r]*qxi[r] + p0i[r]*qxr[r];
            pq1r += p1r[r]*qxr[r] - p1i[r]*qxi[r];
            pq1i += p1r[r]*qxi[r] + p1i[r]*qxr[r];
        }

        float nx0r = a0r*x0r - a0i*x0i + dt*pq0r + f0r;
        float nx0i = a0r*x0i + a0i*x0r + dt*pq0i + f0i;
        float nx1r = a1r*x1r - a1i*x1i + dt*pq1r + f1r;
        float nx1i = a1r*x1i + a1i*x1r + dt*pq1i + f1i;
        x0r = nx0r; x0i = nx0i; x1r = nx1r; x1i = nx1i;

        if (WRITE_X) {
            xp[n0]      = x0r;  xp[64 + n0] = x0i;
            xp[n1]      = x1r;  xp[64 + n1] = x1i;
        }
    }

    if (!WRITE_X) {
        long si = (long)(b*64 + c) * 64;
        ws[OFF_Z + si + n0] = x0r;           ws[OFF_Z + 65536 + si + n0] = x0i;
        ws[OFF_Z + si + n1] = x1r;           ws[OFF_Z + 65536 + si + n1] = x1i;
    }
}

// ---------------------------------------------------------------------------
// boundary chain: x_start[c] = M^64 * x_start[c-1] + Z[c-1]; 1 block / batch
// ---------------------------------------------------------------------------
__global__ void chain_kernel(float* __restrict__ ws)
{
    __shared__ float msr[4096], msi[4096];
    __shared__ float xr[64], xi[64];
    const int b = blockIdx.x, tid = threadIdx.x;   // 64 threads

    for (int i = tid; i < 4096; i += 64) { msr[i] = ws[OFF_MS + i]; msi[i] = ws[OFF_MS + 4096 + i]; }
    xr[tid] = 0.f; xi[tid] = 0.f;
    ws[OFF_XS + (long)(b*64)*64 + tid] = 0.f;
    ws[OFF_XS + 65536 + (long)(b*64)*64 + tid] = 0.f;
    __syncthreads();

    for (int c = 1; c < 64; c++) {
        float ar = 0.f, ai = 0.f;
        for (int j = 0; j < 64; j++) {
            float mr = msr[tid*64 + j], mi = msi[tid*64 + j];
            ar += mr * xr[j] - mi * xi[j];
            ai += mr * xi[j] + mi * xr[j];
        }
        long zidx = (long)(b*64 + (c-1)) * 64 + tid;
        ar += ws[OFF_Z + zidx];
        ai += ws[OFF_Z + 65536 + zidx];
        __syncthreads();
        xr[tid] = ar; xi[tid] = ai;
        long sidx = (long)(b*64 + c) * 64 + tid;
        ws[OFF_XS + sidx] = ar;
        ws[OFF_XS + 65536 + sidx] = ai;
        __syncthreads();
    }
}

// ---------------------------------------------------------------------------
// GEMM2: Y(65536x512) = X(65536x128) @ W2(128x512) + u * D, fp32 WMMA.
// 32x64 strip per wave (2 A frags x 4 B frags).
// ---------------------------------------------------------------------------
__global__ void gemm2_kernel(const float* __restrict__ u, const float* __restrict__ Dv,
                             float* __restrict__ out, const float* __restrict__ ws)
{
    const int lane  = threadIdx.x & 31;
    const int wave  = blockIdx.x * (blockDim.x >> 5) + (threadIdx.x >> 5);
    const int mt    = wave >> 3;          // 32-row strip index (0..2047)
    const int n0    = (wave & 7) * 64;
    const int lr    = lane & 15;
    const int khalf = (lane >> 4) << 1;
    const int hi8   = (lane >> 4) << 3;
    const float* W2 = ws + OFF_W2;
    const float* X  = ws + OFF_X;
    const long r0   = (long)mt * 32;
    const float* xptr0 = X + (r0 + lr) * 128;
    const float* xptr1 = X + (r0 + 16 + lr) * 128;

    v8f a00 = {}, a01 = {}, a02 = {}, a03 = {};
    v8f a10 = {}, a11 = {}, a12 = {}, a13 = {};
    for (int k = 0; k < 128; k += 4) {
        int kk = k + khalf;
        v2f av0 = *(const v2f*)(xptr0 + kk);
        v2f av1 = *(const v2f*)(xptr1 + kk);
        const float* wb = W2 + (long)kk * 512 + 2 * (n0 + lr);
        v2f b0 = *(const v2f*)(wb);
        v2f b1 = *(const v2f*)(wb + 32);
        v2f b2 = *(const v2f*)(wb + 64);
        v2f b3 = *(const v2f*)(wb + 96);
        a00 = __builtin_amdgcn_wmma_f32_16x16x4_f32(false, av0, false, b0, (short)0, a00, false, false);
        a01 = __builtin_amdgcn_wmma_f32_16x16x4_f32(false, av0, false, b1, (short)0, a01, false, false);
        a02 = __builtin_amdgcn_wmma_f32_16x16x4_f32(false, av0, false, b2, (short)0, a02, false, false);
        a03 = __builtin_amdgcn_wmma_f32_16x16x4_f32(false, av0, false, b3, (short)0, a03, false, false);
        a10 = __builtin_amdgcn_wmma_f32_16x16x4_f32(false, av1, false, b0, (short)0, a10, false, false);
        a11 = __builtin_amdgcn_wmma_f32_16x16x4_f32(false, av1, false, b1, (short)0, a11, false, false);
        a12 = __builtin_amdgcn_wmma_f32_16x16x4_f32(false, av1, false, b2, (short)0, a12, false, false);
        a13 = __builtin_amdgcn_wmma_f32_16x16x4_f32(false, av1, false, b3, (short)0, a13, false, false);
    }

    const float d0 = Dv[n0 + lr];
    const float d1 = Dv[n0 + lr + 16];
    const float d2 = Dv[n0 + lr + 32];
    const float d3 = Dv[n0 + lr + 48];
#pragma unroll
    for (int v = 0; v < 8; v++) {
        long row0 = r0 + hi8 + v;
        long row1 = r0 + 16 + hi8 + v;
        const float* up0 = u   + row0 * kD + n0 + lr;
        float*       op0 = out + row0 * kD + n0 + lr;
        const float* up1 = u   + row1 * kD + n0 + lr;
        float*       op1 = out + row1 * kD + n0 + lr;
        op0[0]  = a00[v] + up0[0]  * d0;
        op0[16] = a01[v] + up0[16] * d1;
        op0[32] = a02[v] + up0[32] * d2;
        op0[48] = a03[v] + up0[48] * d3;
        op1[0]  = a10[v] + up1[0]  * d0;
        op1[16] = a11[v] + up1[16] * d1;
        op1[32] = a12[v] + up1[32] * d2;
        op1[48] = a13[v] + up1[48] * d3;
    }
}

// ---------------------------------------------------------------------------
extern "C" void kernel_launch(void* const* d_in, const int* in_sizes, int n_in,
                              void* d_out, int out_size, void* d_ws, size_t ws_size,
                              hipStream_t stream)
{
    const float* u     = (const float*)d_in[0];
    const float* lnr   = (const float*)d_in[1];
    const float* im    = (const float*)d_in[2];
    const float* Pr    = (const float*)d_in[3];
    const float* Pi    = (const float*)d_in[4];
    const float* Qr    = (const float*)d_in[5];
    const float* Qi    = (const float*)d_in[6];
    const float* Br    = (const float*)d_in[7];
    const float* Bi    = (const float*)d_in[8];
    const float* Cr    = (const float*)d_in[9];
    const float* Ci    = (const float*)d_in[10];
    const float* logdt = (const float*)d_in[11];
    const float* Dv    = (const float*)d_in[12];
    float* out = (float*)d_out;
    float* ws  = (float*)d_ws;

    prep_kernel<<<1, 256, 0, stream>>>(lnr, im, Pr, Pi, Qr, Qi, Br, Bi, Cr, Ci, logdt, ws);

    // GEMM1: waves = 2048 strips * 2 col-groups = 4096; 4 waves/block
    gemm1_kernel<<<1024, 128, 0, stream>>>(u, ws);

    // phase 1: local chunk scans from zero state (16 batches * 64 chunks)
    scan_kernel<false><<<1024, 32, 0, stream>>>(lnr, im, Pr, Pi, Qr, Qi, logdt, ws);

    // phase 2: propagate chunk boundary states
    chain_kernel<<<16, 64, 0, stream>>>(ws);

    // phase 3: final scans writing full state trajectory in place
    scan_kernel<true><<<1024, 32, 0, stream>>>(lnr, im, Pr, Pi, Qr, Qi, logdt, ws);

    // GEMM2: waves = 2048 strips * 8 col-groups = 16384; 4 waves/block
    gemm2_kernel<<<4096, 128, 0, stream>>>(u, Dv, out, ws);
}